// ChannelsShuffle_42339787603994
// MI455X (gfx1250) — compile-verified
//
#include <hip/hip_runtime.h>
#include <hip/hip_bf16.h>
#include <stdint.h>
#include <stddef.h>

// ---------------------------------------------------------------------------
// ChannelsShuffle on MI455X (gfx1250)
//   X: (32, 128, 16384) fp32.  out[b,i,:] = X[b, perm[b,i], :]
//   perm derived from JAX threefry randomness (key 42) -> computed on device.
//   Roofline: 512 MB total traffic / 23.3 TB/s  ->  ~22 us.  Pure bandwidth.
// ---------------------------------------------------------------------------

#define BATCH 32
#define CH    128
#define LEN   16384

typedef float vf4 __attribute__((ext_vector_type(4)));
typedef int   v4i __attribute__((vector_size(16)));   // matches builtin sig

#ifndef __has_builtin
#define __has_builtin(x) 0
#endif

#if defined(__HIP_DEVICE_COMPILE__) &&                                   \
    __has_builtin(__builtin_amdgcn_global_load_async_to_lds_b128) &&     \
    __has_builtin(__builtin_amdgcn_s_wait_asynccnt)
#define ASYNC_OK 1
typedef __attribute__((address_space(1))) v4i* gv4i_ptr;  // global (AS1)
typedef __attribute__((address_space(3))) v4i* lv4i_ptr;  // LDS (AS3)
#else
#define ASYNC_OK 0
#endif

// ----------------------------- threefry2x32 --------------------------------
__device__ __forceinline__ unsigned rotl32(unsigned v, int n) {
  return (v << n) | (v >> (32 - n));
}

__device__ __forceinline__ void threefry2x32(unsigned k0, unsigned k1,
                                             unsigned& x0, unsigned& x1) {
  const unsigned ks0 = k0, ks1 = k1, ks2 = k0 ^ k1 ^ 0x1BD11BDAu;
  const int ra[4] = {13, 15, 26, 6};
  const int rb[4] = {17, 29, 16, 24};
  x0 += ks0; x1 += ks1;
#pragma unroll
  for (int i = 0; i < 4; ++i) { x0 += x1; x1 = rotl32(x1, ra[i]); x1 ^= x0; }
  x0 += ks1; x1 += ks2 + 1u;
#pragma unroll
  for (int i = 0; i < 4; ++i) { x0 += x1; x1 = rotl32(x1, rb[i]); x1 ^= x0; }
  x0 += ks2; x1 += ks0 + 2u;
#pragma unroll
  for (int i = 0; i < 4; ++i) { x0 += x1; x1 = rotl32(x1, ra[i]); x1 ^= x0; }
  x0 += ks0; x1 += ks1 + 3u;
#pragma unroll
  for (int i = 0; i < 4; ++i) { x0 += x1; x1 = rotl32(x1, rb[i]); x1 ^= x0; }
  x0 += ks1; x1 += ks2 + 4u;
#pragma unroll
  for (int i = 0; i < 4; ++i) { x0 += x1; x1 = rotl32(x1, ra[i]); x1 ^= x0; }
  x0 += ks2; x1 += ks0 + 5u;
}

// JAX random_bits for a 4096-element draw: counts = iota(4096) split in half,
// pairs (i, i+2048) -> (y0, y1); out = concat(y0, y1).
__device__ __forceinline__ unsigned jax_bits_4096(unsigned k0, unsigned k1,
                                                  int idx) {
  const int half = (BATCH * CH) / 2;  // 2048
  unsigned x0, x1;
  if (idx < half) { x0 = (unsigned)idx;          x1 = (unsigned)(idx + half); }
  else            { x0 = (unsigned)(idx - half); x1 = (unsigned)idx; }
  threefry2x32(k0, k1, x0, x1);
  return (idx < half) ? x0 : x1;
}

__device__ __forceinline__ float bits_to_u01(unsigned b) {
  return __uint_as_float((b >> 9) | 0x3f800000u) - 1.0f;
}

// ------------------------- perm computation --------------------------------
// One block per batch, 128 threads (one per channel). Writes perm to d_ws.
__global__ __launch_bounds__(CH) void ChannelsShuffle_perm_kernel(
    int* __restrict__ perm_out) {
  __shared__ float keyv[CH];
  __shared__ int   maskv[CH];
  __shared__ int   order[CH];
  __shared__ int   scan[CH];

  const int b = blockIdx.x;
  const int c = threadIdx.x;

  // split(key(42)): threefry(key=(0,42), counts=iota(4)) -> k1=(y0_0,y0_1),
  // k2=(y1_0,y1_1) where pair i is (i, i+2).
  unsigned a0 = 0u, a1 = 2u; threefry2x32(0u, 42u, a0, a1);  // pair 0
  unsigned c0 = 1u, c1 = 3u; threefry2x32(0u, 42u, c0, c1);  // pair 1
  const unsigned k1_0 = a0, k1_1 = c0;
  const unsigned k2_0 = a1, k2_1 = c1;

  const int idx = b * CH + c;
  const float u_mask = bits_to_u01(jax_bits_4096(k1_0, k1_1, idx));
  const float u_r    = bits_to_u01(jax_bits_4096(k2_0, k2_1, idx));
  const int m = (u_mask < 0.5f) ? 1 : 0;  // bernoulli(1 - p_shuffle)

  keyv[c]  = m ? u_r : 2.0f;  // where(mask, r, 2.0)
  maskv[c] = m;
  scan[c]  = m;
  __syncthreads();

  // Stable argsort via rank counting (tie-break by index == stable).
  const float kv = keyv[c];
  int rank = 0;
#pragma unroll 8
  for (int j = 0; j < CH; ++j) {
    const float kj = keyv[j];
    rank += (kj < kv) || (kj == kv && j < c);
  }
  order[rank] = c;

  // Inclusive scan of mask (Hillis-Steele) -> cumsum.
  for (int off = 1; off < CH; off <<= 1) {
    __syncthreads();
    const int v = (c >= off) ? scan[c - off] : 0;
    __syncthreads();
    scan[c] += v;
  }
  __syncthreads();

  int pr = scan[c] - 1;
  pr = pr < 0 ? 0 : (pr > CH - 1 ? CH - 1 : pr);
  perm_out[idx] = maskv[c] ? order[pr] : c;
}

// ------------------------------ row gather ---------------------------------
// One block per output row (64 KB). 256 threads, 16 b128 elements per thread.
// Async Global->LDS DMA (ASYNCcnt), double-buffered; drain via ds_load_b128 +
// non-temporal global_store_b128 (512 MB working set > 192 MB L2 -> stream).
__global__ __launch_bounds__(256) void ChannelsShuffle_gather_kernel(
    const float* __restrict__ X, float* __restrict__ out,
    const int* __restrict__ perm) {
  const int o   = blockIdx.x;      // row id = b*CH + i
  const int b   = o >> 7;
  const int tid = threadIdx.x;
  const int p   = perm[o];

  const vf4* __restrict__ src =
      (const vf4*)(X + (size_t)(b * CH + p) * (size_t)LEN);
  vf4* __restrict__ dst = (vf4*)(out + (size_t)o * (size_t)LEN);

  constexpr int NV    = LEN / 4;   // 4096 vf4 per row
  constexpr int STEPS = NV / 256;  // 16 tiles of 4 KB

#if ASYNC_OK
  __shared__ vf4 buf[2][256];      // 8 KB double buffer

  // Prologue: tile 0 in flight.
  __builtin_amdgcn_global_load_async_to_lds_b128(
      (gv4i_ptr)(src + tid), (lv4i_ptr)&buf[0][tid], 0, 0);

#pragma unroll
  for (int t = 0; t < STEPS; ++t) {
    if (t + 1 < STEPS) {
      __builtin_amdgcn_global_load_async_to_lds_b128(
          (gv4i_ptr)(src + (t + 1) * 256 + tid),
          (lv4i_ptr)&buf[(t + 1) & 1][tid], 0, 0);
      __builtin_amdgcn_s_wait_asynccnt(1);  // tile t has landed in LDS
    } else {
      __builtin_amdgcn_s_wait_asynccnt(0);
    }
    asm volatile("" ::: "memory");          // keep ds read after the wait
    const vf4 v = buf[t & 1][tid];          // ds_load_b128 (own lane's slot)
    __builtin_nontemporal_store(v, dst + t * 256 + tid);
  }
#else
#pragma unroll
  for (int t = 0; t < STEPS; ++t) {
    const int e = t * 256 + tid;
    const vf4 v = __builtin_nontemporal_load(src + e);
    __builtin_nontemporal_store(v, dst + e);
  }
#endif
}

// ------------------------------- launcher ----------------------------------
extern "C" void kernel_launch(void* const* d_in, const int* in_sizes, int n_in,
                              void* d_out, int out_size, void* d_ws,
                              size_t ws_size, hipStream_t stream) {
  (void)in_sizes; (void)n_in; (void)out_size; (void)ws_size;
  const float* X   = (const float*)d_in[0];
  float*       out = (float*)d_out;
  int*         perm = (int*)d_ws;  // 32*128 ints = 16 KB scratch

  ChannelsShuffle_perm_kernel<<<BATCH, CH, 0, stream>>>(perm);
  ChannelsShuffle_gather_kernel<<<BATCH * CH, 256, 0, stream>>>(X, out, perm);
}